// MultiHeadAttention_58153857188091
// MI455X (gfx1250) — compile-verified
//
#include <hip/hip_runtime.h>
#include <cmath>

// Problem dims (fixed by reference)
#define BN   256   // batch
#define NS   64    // sequence length / num weight blocks
#define DM   512   // d_model
#define HD   512   // n_head * d_k
#define NH   8
#define DKV  64
#define EPSV 1e-7f

typedef float v2f __attribute__((ext_vector_type(2)));
typedef float v8f __attribute__((ext_vector_type(8)));
typedef unsigned int u32x4 __attribute__((ext_vector_type(4)));
typedef int i32x4 __attribute__((ext_vector_type(4)));
typedef int i32x8 __attribute__((ext_vector_type(8)));

// CDNA5 fp32 WMMA: D(16x16) = A(16x4) * B(4x16) + C, exact fp32.
__device__ __forceinline__ v8f wmma4(v2f a, v2f b, v8f c) {
  return __builtin_amdgcn_wmma_f32_16x16x4_f32(false, a, false, b, (short)0, c,
                                               false, false);
}

// LDS tile row stride in floats: 64 data + 4 pad DWORDs (TDM pad_interval=64dw,
// pad_amount=4dw) -> bank-conflict-spreading stride 68.
#define TROW 68

// Issue a TDM DMA of W[n][kt..kt+63][o0..o0+63] (row stride HD floats) into
// LDS at lds_byte_addr, padded to TROW floats per row. D# per ISA 08 §8.
// This toolchain's builtin takes 6 args: (g0, g1, g2, g3, extra, cpol).
__device__ __forceinline__ void tdm_load_wtile(const float* __restrict__ W,
                                               int n, int kt, int o0,
                                               unsigned int lds_byte_addr) {
  unsigned long long ga =
      (unsigned long long)(uintptr_t)(W + ((size_t)n * DM + kt) * HD + o0);
  u32x4 g0;
  g0.x = 1u;                                   // count=1, user descriptor
  g0.y = lds_byte_addr;                        // lds_addr (bytes)
  g0.z = (unsigned int)ga;                     // global_addr[31:0]
  g0.w = (unsigned int)((ga >> 32) & 0x1FFFFFFu) | 0x80000000u; // [56:32]|type=2
  i32x8 g1;
  g1[0] = (int)((2u << 16)      // data_size = 4 bytes
              | (1u << 20)      // pad_enable
              | (5u << 22)      // pad_interval: 64 DWORDs
              | (3u << 25));    // pad_amount: 4 DWORDs
  g1[1] = (int)(64u << 16);     // tensor_dim0 = 64 (bits [79:48] lo16)
  g1[2] = (int)(64u << 16);     // tensor_dim0 hi = 0 | tensor_dim1 = 64 (lo16)
  g1[3] = (int)(64u << 16);     // tensor_dim1 hi = 0 | tile_dim0 = 64
  g1[4] = 64;                   // tile_dim1 = 64, tile_dim2 = 0
  g1[5] = 512;                  // tensor_dim0_stride = HD (lo32)
  g1[6] = 0;                    // stride0 hi16 | tensor_dim1_stride lo16
  g1[7] = 0;                    // tensor_dim1_stride hi32
  i32x4 gz  = {0, 0, 0, 0};               // groups 2/3 unused (2-D tensor)
  i32x8 gz8 = {0, 0, 0, 0, 0, 0, 0, 0};   // extra group (unused)
  __builtin_amdgcn_tensor_load_to_lds(g0, g1, gz, gz, gz8, 0);
}

// ---------------------------------------------------------------------------
// Kernel 1: group linear  Y[b, n, :] = X[b, n, :] @ W[n]   (per-position GEMM)
// grid = (HD/64, BN/128, NS), block = 256 (8 waves). Wave computes 16x64 strip.
// W tiles DMA'd into LDS by the Tensor Data Mover, double-buffered so the DMA
// of tile t+1 overlaps the 64 WMMAs of tile t. TENSORcnt orders the pipeline.
// ---------------------------------------------------------------------------
__global__ void group_linear(const float* __restrict__ X,
                             const float* __restrict__ W,
                             float* __restrict__ Y) {
  __shared__ __align__(16) float ldsB[2][64 * TROW];
  const int n  = blockIdx.z;
  const int o0 = blockIdx.x * 64;
  const int m0 = blockIdx.y * 128;
  const int tid  = threadIdx.x;
  const int wave = tid >> 5;
  const int lane = tid & 31;
  const int hi   = lane >> 4;
  const int ln   = lane & 15;

  const int rowA = m0 + wave * 16 + ln;                      // A-frag row (m)
  const float* Arow = X + ((size_t)rowA * NS + n) * DM;      // + k

  const unsigned int ldsAddr0 = (unsigned int)(uintptr_t)&ldsB[0][0];
  const unsigned int ldsAddr1 = (unsigned int)(uintptr_t)&ldsB[1][0];

  if (wave == 0) tdm_load_wtile(W, n, 0, o0, ldsAddr0);      // prime buffer 0

  v8f acc[4] = {};
  for (int it = 0; it < DM / 64; ++it) {
    const int kt  = it * 64;
    const int buf = it & 1;
    __syncthreads();   // prior readers of buf^1 done -> safe to DMA over it
    if (wave == 0) {
      if (it < DM / 64 - 1) {
        tdm_load_wtile(W, n, kt + 64, o0, buf ? ldsAddr0 : ldsAddr1);
        __builtin_amdgcn_s_wait_tensorcnt(1);  // tile 'it' landed (in-order)
      } else {
        __builtin_amdgcn_s_wait_tensorcnt(0);
      }
    }
    __syncthreads();   // publish tile to all waves

    const float* Bt = &ldsB[buf][0];
#pragma unroll
    for (int ks = 0; ks < 64; ks += 4) {
      const int k = ks + 2 * hi;
      v2f a = *(const v2f*)(Arow + kt + k);
#pragma unroll
      for (int tj = 0; tj < 4; ++tj) {
        const int nn = tj * 16 + ln;
        v2f b;
        b.x = Bt[k * TROW + nn];         // B[k][n]   = W[kt+k][o0+n]
        b.y = Bt[(k + 1) * TROW + nn];   // B[k+1][n]
        acc[tj] = wmma4(a, b, acc[tj]);
      }
    }
  }
  // store: Y is (B, N, H)
#pragma unroll
  for (int tj = 0; tj < 4; ++tj)
#pragma unroll
    for (int r = 0; r < 8; ++r) {
      int m = m0 + wave * 16 + r + 8 * hi;
      int o = o0 + tj * 16 + ln;
      Y[((size_t)m * NS + n) * HD + o] = acc[tj][r];
    }
}

// ---------------------------------------------------------------------------
// Kernel 2: logits S[h,b,q,k] = (Qh . Kh) / sqrt(64), written to d_out attn
// region in the reference's (h, b, q, k) layout. grid = (BN, 4 q-tiles),
// block = 256 (wave w == head w), wave computes 16x64 logit strip.
// ---------------------------------------------------------------------------
__global__ void attn_scores(const float* __restrict__ qh,
                            const float* __restrict__ kh,
                            float* __restrict__ S) {
  const int b  = blockIdx.x;
  const int qt = blockIdx.y;
  const int h    = threadIdx.x >> 5;
  const int lane = threadIdx.x & 31;
  const int hi   = lane >> 4;
  const int ln   = lane & 15;

  const int q = qt * 16 + ln;
  const float* Q = qh + ((size_t)b * NS + q) * HD + h * DKV;

  v8f acc[4] = {};
#pragma unroll
  for (int ks = 0; ks < DKV; ks += 4) {
    const int k = ks + 2 * hi;
    v2f a = *(const v2f*)(Q + k);
#pragma unroll
    for (int tj = 0; tj < 4; ++tj) {
      const int key = tj * 16 + ln;
      v2f bb = *(const v2f*)(kh + ((size_t)b * NS + key) * HD + h * DKV + k);
      acc[tj] = wmma4(a, bb, acc[tj]);
    }
  }
  const float scale = 0.125f;  // 1/sqrt(64)
  float* Sb = S + ((size_t)h * BN + b) * NS * NS;
#pragma unroll
  for (int tj = 0; tj < 4; ++tj)
#pragma unroll
    for (int r = 0; r < 8; ++r) {
      int qq = qt * 16 + r + 8 * hi;
      int kk = tj * 16 + ln;
      Sb[qq * NS + kk] = acc[tj][r] * scale;
    }
}

// ---------------------------------------------------------------------------
// Kernel 3: per-row softmax + top-k(5) sparsify + renorm, in place on the
// d_out attn region. One thread per row (64 elems), row cached in padded LDS.
// Tie-aware 5th-largest: v5 = min{ x_i : #{j : x_j > x_i} < 5 }.
// ---------------------------------------------------------------------------
__global__ void softmax_topk(float* __restrict__ attn) {
  __shared__ float lds[128 * 65];
  const int row = blockIdx.x * 128 + threadIdx.x;   // 131072 rows total
  float* rp = attn + (size_t)row * NS;
  float* l  = &lds[threadIdx.x * 65];

  float m = -INFINITY;
  for (int j = 0; j < NS; ++j) { float x = rp[j]; l[j] = x; m = fmaxf(m, x); }
  float s = 0.f;
  for (int j = 0; j < NS; ++j) { float e = expf(l[j] - m); l[j] = e; s += e; }
  float inv = 1.f / s;
  for (int j = 0; j < NS; ++j) l[j] *= inv;

  float v5 = INFINITY;
  for (int i = 0; i < NS; ++i) {
    float xi = l[i];
    int cg = 0;
    for (int j = 0; j < NS; ++j) cg += (l[j] > xi) ? 1 : 0;
    if (cg < 5) v5 = fminf(v5, xi);
  }
  const float delta = v5 + EPSV;
  float sw = 0.f;
  for (int j = 0; j < NS; ++j) { float w = fmaxf(l[j] - delta, 0.f); l[j] = w; sw += w; }
  const float invw = 1.f / (sw + EPSV);
  for (int j = 0; j < NS; ++j) rp[j] = l[j] * invw;
}

// ---------------------------------------------------------------------------
// Kernel 4: context  ctx[b,q,h*64+d] = sum_k P[h,b,q,k] * vh[b,k,h*64+d]
// grid = (BN, 4), block = 256 (wave == head).
// ---------------------------------------------------------------------------
__global__ void attn_ctx(const float* __restrict__ P,
                         const float* __restrict__ vh,
                         float* __restrict__ ctx) {
  const int b  = blockIdx.x;
  const int qt = blockIdx.y;
  const int h    = threadIdx.x >> 5;
  const int lane = threadIdx.x & 31;
  const int hi   = lane >> 4;
  const int ln   = lane & 15;

  const int q = qt * 16 + ln;
  const float* Pr = P + (((size_t)h * BN + b) * NS + q) * NS;

  v8f acc[4] = {};
#pragma unroll
  for (int ks = 0; ks < NS; ks += 4) {
    const int k = ks + 2 * hi;
    v2f a = *(const v2f*)(Pr + k);
#pragma unroll
    for (int tj = 0; tj < 4; ++tj) {
      const int d = tj * 16 + ln;
      const float* Vp = vh + ((size_t)b * NS + k) * HD + h * DKV + d;
      v2f bb;
      bb.x = Vp[0];        // B[k][d]
      bb.y = Vp[HD];       // B[k+1][d]
      acc[tj] = wmma4(a, bb, acc[tj]);
    }
  }
#pragma unroll
  for (int tj = 0; tj < 4; ++tj)
#pragma unroll
    for (int r = 0; r < 8; ++r) {
      int qq = qt * 16 + r + 8 * hi;
      int d  = h * DKV + tj * 16 + ln;
      ctx[((size_t)b * NS + qq) * HD + d] = acc[tj][r];
    }
}

// ---------------------------------------------------------------------------
// Kernel 5: out = sigmoid(ctx@gWᵀ+gb) * tanh(ctx@fcWᵀ+fcb)
// grid = (8 col-tiles, 128), block = 256 (8 waves); wave = 16 rows x 64 cols,
// dual accumulators share the A fragments. W rows contiguous in k -> direct
// v2f B-frag loads. Also emits the scalar third output.
// ---------------------------------------------------------------------------
__global__ void gated_out(const float* __restrict__ ctx,
                          const float* __restrict__ fcW,
                          const float* __restrict__ fcB,
                          const float* __restrict__ gW,
                          const float* __restrict__ gB,
                          float* __restrict__ out,
                          float* __restrict__ scalarOut) {
  const int oc   = blockIdx.x * 64;
  const int wave = threadIdx.x >> 5;
  const int lane = threadIdx.x & 31;
  const int hi   = lane >> 4;
  const int ln   = lane & 15;
  const int rowBase = (blockIdx.y * 8 + wave) * 16;

  const float* A = ctx + (size_t)(rowBase + ln) * HD;

  v8f accF[4] = {}, accG[4] = {};
#pragma unroll 4
  for (int ks = 0; ks < HD; ks += 4) {
    const int k = ks + 2 * hi;
    v2f a = *(const v2f*)(A + k);
#pragma unroll
    for (int tj = 0; tj < 4; ++tj) {
      const int o = oc + tj * 16 + ln;
      v2f bf = *(const v2f*)(fcW + (size_t)o * HD + k);
      v2f bg = *(const v2f*)(gW  + (size_t)o * HD + k);
      accF[tj] = wmma4(a, bf, accF[tj]);
      accG[tj] = wmma4(a, bg, accG[tj]);
    }
  }
#pragma unroll
  for (int tj = 0; tj < 4; ++tj)
#pragma unroll
    for (int r = 0; r < 8; ++r) {
      int rr = rowBase + r + 8 * hi;
      int o  = oc + tj * 16 + ln;
      float fc = accF[tj][r] + fcB[o];
      float gt = accG[tj][r] + gB[o];
      out[(size_t)rr * HD + o] = tanhf(fc) * (1.f / (1.f + expf(-gt)));
    }
  if (blockIdx.x == 0 && blockIdx.y == 0 && threadIdx.x == 0)
    scalarOut[0] = 0.0f;
}

// ---------------------------------------------------------------------------
extern "C" void kernel_launch(void* const* d_in, const int* in_sizes, int n_in,
                              void* d_out, int out_size, void* d_ws, size_t ws_size,
                              hipStream_t stream) {
  const float* q      = (const float*)d_in[0];
  const float* k      = (const float*)d_in[1];
  const float* v      = (const float*)d_in[2];
  const float* Wq     = (const float*)d_in[3];
  const float* Wk     = (const float*)d_in[4];
  const float* Wv     = (const float*)d_in[5];
  const float* fc_w   = (const float*)d_in[6];
  const float* fc_b   = (const float*)d_in[7];
  const float* gate_w = (const float*)d_in[8];
  const float* gate_b = (const float*)d_in[9];

  const size_t ACT = (size_t)BN * NS * HD;        // 8,388,608 elements
  float* ws  = (float*)d_ws;
  float* qh  = ws;
  float* kh  = ws + ACT;
  float* vh  = ws + 2 * ACT;
  float* ctx = ws + 3 * ACT;

  float* out    = (float*)d_out;
  float* attn   = out + ACT;            // (h, b, q, k) region of d_out
  float* scalar = out + 2 * ACT;        // trailing scalar output

  dim3 blk(256);

  dim3 g1(HD / 64, BN / 128, NS);
  group_linear<<<g1, blk, 0, stream>>>(q, Wq, qh);
  group_linear<<<g1, blk, 0, stream>>>(k, Wk, kh);
  group_linear<<<g1, blk, 0, stream>>>(v, Wv, vh);

  attn_scores<<<dim3(BN, 4), blk, 0, stream>>>(qh, kh, attn);
  softmax_topk<<<dim3((BN * NH * NS) / 128), dim3(128), 0, stream>>>(attn);
  attn_ctx<<<dim3(BN, 4), blk, 0, stream>>>(attn, vh, ctx);

  gated_out<<<dim3(HD / 64, (BN * NS) / (8 * 16)), blk, 0, stream>>>(
      ctx, fc_w, fc_b, gate_w, gate_b, out, scalar);
}